// PCSA_34102040330780
// MI455X (gfx1250) — compile-verified
//
#include <hip/hip_runtime.h>
#include <hip/hip_bf16.h>

// PCSA for MI455X (gfx1250, wave32).
// Memory-bound problem: 2 passes over x (128 MB) + ~10.5 MB scratch.
// WMMA f32 16x16x4 for the per-head channel attention GEMMs,
// GLOBAL_LOAD_ASYNC_TO_LDS_B128 for the LDS-staged reduction pass.

#define BATCH 32
#define CHN   256
#define HH    64
#define WW    64
#define PLANES (BATCH * CHN)        // 8192
#define NPOS  64                    // ph*pw = 8*8
#define NHEADS 8
#define HD    32

typedef __attribute__((ext_vector_type(2))) float v2f;
typedef __attribute__((ext_vector_type(8))) float v8f;

__device__ __forceinline__ v8f wmma_f32_16x16x4(v2f a, v2f b, v8f c) {
  // D = A(16x4 f32) * B(4x16 f32) + C(16x16 f32)
  return __builtin_amdgcn_wmma_f32_16x16x4_f32(
      /*neg_a=*/false, a, /*neg_b=*/false, b,
      /*c_mod=*/(short)0, c, /*reuse_a=*/false, /*reuse_b=*/false);
}

__device__ __forceinline__ float sigmoidf_(float x) {
  return 1.0f / (1.0f + __expf(-x));
}

// Raw LDS byte address (relative to wave LDS base) of a shared-memory pointer.
__device__ __forceinline__ unsigned lds_addr_of(const void* p) {
  return (unsigned)(unsigned long long)(
      (__attribute__((address_space(3))) const char*)p);
}

// Async copy 16 bytes global -> LDS (per lane), tracked by ASYNCcnt.
__device__ __forceinline__ void async_copy_b128(unsigned lds_byte_addr,
                                                const void* gaddr) {
  asm volatile("global_load_async_to_lds_b128 %0, %1, off"
               :
               : "v"(lds_byte_addr), "v"(gaddr)
               : "memory");
}

__device__ __forceinline__ void async_wait0() {
  asm volatile("s_wait_asynccnt 0" ::: "memory");
}

// ---------------------------------------------------------------------------
// Kernel 1: per-(b,c) plane: row means (over W), col means (over H),
// 8x8 pooled means. One pass over x, staged through LDS with async copies.
// ---------------------------------------------------------------------------
__global__ __launch_bounds__(256)
void reduce_kernel(const float* __restrict__ x,
                   float* __restrict__ rowMean,   // (B,C,H)
                   float* __restrict__ colMean,   // (B,C,W)
                   float* __restrict__ pooled) {  // (B,C,8,8) -> 64
  __shared__ float tile[HH * WW];   // 16 KB
  __shared__ float red[256];
  const int plane = blockIdx.x;
  const int t = threadIdx.x;
  const float4* __restrict__ p4 = (const float4*)(x + (size_t)plane * (HH * WW));

  // ---- async global -> LDS staging (no VGPR round-trip)
  {
    const unsigned base = lds_addr_of(tile);
#pragma unroll
    for (int i = 0; i < 4; ++i) {
      const int idx4 = i * 256 + t;
      async_copy_b128(base + (unsigned)idx4 * 16u, p4 + idx4);
    }
    async_wait0();
  }
  __syncthreads();

  // ---- row partials: thread t -> row r = t>>2, quarter q = t&3
  {
    const int r = t >> 2, q = t & 3;
    float s = 0.f;
#pragma unroll
    for (int j = 0; j < 16; ++j) s += tile[r * WW + q * 16 + j];
    red[t] = s;
  }
  __syncthreads();
  if (t < HH) {
    float s = red[t * 4] + red[t * 4 + 1] + red[t * 4 + 2] + red[t * 4 + 3];
    rowMean[(size_t)plane * HH + t] = s * (1.0f / (float)WW);
  }
  __syncthreads();

  // ---- col partials: thread t -> col c = t&63, row-seg g = t>>6 (16 rows)
  {
    const int c = t & 63, g = t >> 6;
    float s = 0.f;
#pragma unroll
    for (int j = 0; j < 16; ++j) s += tile[(g * 16 + j) * WW + c];
    red[t] = s;
  }
  __syncthreads();
  if (t < WW) {
    float s = red[t] + red[64 + t] + red[128 + t] + red[192 + t];
    colMean[(size_t)plane * WW + t] = s * (1.0f / (float)HH);
  }

  // ---- 8x8 patch means (uses only `tile`)
  if (t < 64) {
    const int ph = t >> 3, pw = t & 7;
    float s = 0.f;
#pragma unroll
    for (int i = 0; i < 8; ++i)
#pragma unroll
      for (int j = 0; j < 8; ++j) s += tile[(ph * 8 + i) * WW + (pw * 8 + j)];
    pooled[(size_t)plane * 64 + t] = s * (1.0f / 64.0f);
  }
}

// ---------------------------------------------------------------------------
// Kernel 2: strip attention. grid = (B, 2). dir 0: rowMean->h_att (gnh),
// dir 1: colMean->w_att (gnw). 256 threads = 256 channels, L = 64.
// ---------------------------------------------------------------------------
__global__ __launch_bounds__(256)
void strip_kernel(const float* __restrict__ rowMean,
                  const float* __restrict__ colMean,
                  const float* __restrict__ dw1, const float* __restrict__ db1,
                  const float* __restrict__ dw2, const float* __restrict__ db2,
                  const float* __restrict__ dw3, const float* __restrict__ db3,
                  const float* __restrict__ dw4, const float* __restrict__ db4,
                  const float* __restrict__ gnh_w, const float* __restrict__ gnh_b,
                  const float* __restrict__ gnw_w, const float* __restrict__ gnw_b,
                  float* __restrict__ h_att, float* __restrict__ w_att) {
  __shared__ float feat[CHN * 64];   // 64 KB
  __shared__ float rsum[256];
  __shared__ float rsq[256];

  const int b = blockIdx.x;
  const int dir = blockIdx.y;
  const int c = threadIdx.x;
  const int g = c >> 6;            // group 0..3
  const int lc = c & 63;           // channel within group
  const int k = 3 + 2 * g;         // kernel size 3,5,7,9
  const int kh = k >> 1;

  const float* __restrict__ in = (dir == 0) ? rowMean : colMean;
  const float* __restrict__ gw = (dir == 0) ? gnh_w : gnw_w;
  const float* __restrict__ gb = (dir == 0) ? gnh_b : gnw_b;
  float* __restrict__ outp = (dir == 0) ? h_att : w_att;

  const float* dwp = (g == 0) ? dw1 : (g == 1) ? dw2 : (g == 2) ? dw3 : dw4;
  const float* dbp = (g == 0) ? db1 : (g == 1) ? db2 : (g == 2) ? db3 : db4;

  float w[9];
  for (int j = 0; j < 9; ++j) w[j] = (j < k) ? dwp[lc * k + j] : 0.f;
  const float bias = dbp[lc];

  float s[64];
  const float* src = in + ((size_t)b * CHN + c) * 64;
#pragma unroll
  for (int l = 0; l < 64; ++l) s[l] = src[l];

  float s0 = 0.f, s1 = 0.f;
  for (int l = 0; l < 64; ++l) {
    float acc = bias;
    for (int j = 0; j < 9; ++j) {
      if (j < k) {
        int p = l + j - kh;
        if (p >= 0 && p < 64) acc += w[j] * s[p];
      }
    }
    feat[c * 64 + l] = acc;
    s0 += acc;
    s1 += acc * acc;
  }
  rsum[c] = s0;
  rsq[c] = s1;
  __syncthreads();

  // group reduction over 64 channels (threads g*64..g*64+63)
  for (int off = 32; off > 0; off >>= 1) {
    if ((c & 63) < off) {
      rsum[c] += rsum[c + off];
      rsq[c] += rsq[c + off];
    }
    __syncthreads();
  }
  const float m = rsum[g * 64] * (1.0f / 4096.0f);
  const float var = rsq[g * 64] * (1.0f / 4096.0f) - m * m;
  const float rs = rsqrtf(var + 1e-5f);
  const float cw = gw[c], cb = gb[c];

  float* dst = outp + ((size_t)b * CHN + c) * 64;
#pragma unroll
  for (int l = 0; l < 64; ++l) {
    float v = (feat[c * 64 + l] - m) * rs * cw + cb;
    dst[l] = sigmoidf_(v);
  }
}

// ---------------------------------------------------------------------------
// Kernel 3: pooled channel self-attention. One block per batch, 8 waves,
// wave h handles head h with V_WMMA_F32_16X16X4_F32.
// ---------------------------------------------------------------------------
__global__ __launch_bounds__(256)
void pool_att_kernel(const float* __restrict__ pooled,
                     const float* __restrict__ norm_w,
                     const float* __restrict__ norm_b,
                     const float* __restrict__ wq,
                     const float* __restrict__ wk,
                     const float* __restrict__ wv,
                     float* __restrict__ c_att) {
  __shared__ float yln[CHN * NPOS];       // 64 KB normalized pooled features
  __shared__ float S[NHEADS][HD * HD];    // 32 KB attention scores
  __shared__ float osum[NHEADS][HD];      // 1 KB per-row output sums
  __shared__ float red[512];

  const int b = blockIdx.x;
  const int t = threadIdx.x;
  const int h = t >> 5;          // wave id == head
  const int lane = t & 31;
  const int lm = lane & 15;
  const int hi = lane >> 4;

  ((float*)osum)[t] = 0.f;

  // ---- load + stats (coalesced)
  const float* yb = pooled + (size_t)b * CHN * NPOS;
  float s0 = 0.f, s1 = 0.f;
#pragma unroll 4
  for (int i = 0; i < 64; ++i) {
    int idx = i * 256 + t;
    float v = yb[idx];
    yln[idx] = v;
    s0 += v;
    s1 += v * v;
  }
  red[t] = s0;
  red[256 + t] = s1;
  __syncthreads();
  for (int off = 128; off > 0; off >>= 1) {
    if (t < off) {
      red[t] += red[t + off];
      red[256 + t] += red[256 + t + off];
    }
    __syncthreads();
  }
  const float m = red[0] * (1.0f / 16384.0f);
  const float var = red[256] * (1.0f / 16384.0f) - m * m;
  const float rs = rsqrtf(var + 1e-5f);
  __syncthreads();
#pragma unroll 4
  for (int i = 0; i < 64; ++i) {
    int idx = i * 256 + t;
    int c = idx >> 6;
    yln[idx] = (yln[idx] - m) * rs * norm_w[c] + norm_b[c];
  }
  __syncthreads();

  // ---- S = scale * Q K^T   (per head: 32x32, K-dim = 64)
  const int ch0 = h * HD;
  const float scale = 0.17677669529663687f;   // 32^-0.5
#pragma unroll
  for (int ti = 0; ti < 2; ++ti) {
#pragma unroll
    for (int tj = 0; tj < 2; ++tj) {
      v8f acc = {};
      const int dch = ch0 + ti * 16 + lm;   // A row channel (q)
      const int ech = ch0 + tj * 16 + lm;   // B col channel (k)
      const float qs = wq[dch];
      const float ks = wk[ech];
#pragma unroll
      for (int kk = 0; kk < 16; ++kk) {
        const int n0 = kk * 4 + 2 * hi;
        v2f a, bm;
        a.x = yln[dch * 64 + n0] * qs;
        a.y = yln[dch * 64 + n0 + 1] * qs;
        bm.x = yln[ech * 64 + n0] * ks;
        bm.y = yln[ech * 64 + n0 + 1] * ks;
        acc = wmma_f32_16x16x4(a, bm, acc);
      }
#pragma unroll
      for (int v = 0; v < 8; ++v) {
        const int d = ti * 16 + v + 8 * hi;
        const int e = tj * 16 + lm;
        S[h][d * HD + e] = acc[v] * scale;
      }
    }
  }
  __syncthreads();

  // ---- softmax over e, lane == row
  {
    float* Sh = S[h];
    const int d = lane;
    float mx = -1e30f;
#pragma unroll
    for (int e = 0; e < HD; ++e) mx = fmaxf(mx, Sh[d * HD + e]);
    float sum = 0.f;
#pragma unroll
    for (int e = 0; e < HD; ++e) {
      float ex = __expf(Sh[d * HD + e] - mx);
      Sh[d * HD + e] = ex;
      sum += ex;
    }
    const float inv = 1.0f / sum;
#pragma unroll
    for (int e = 0; e < HD; ++e) Sh[d * HD + e] *= inv;
  }
  __syncthreads();

  // ---- O = att * V, accumulate row sums directly (mean over n)
  {
    float* Sh = S[h];
#pragma unroll
    for (int ti = 0; ti < 2; ++ti) {
#pragma unroll
      for (int tj = 0; tj < 4; ++tj) {
        v8f acc = {};
        const int arow = ti * 16 + lm;
        const int n = tj * 16 + lm;
#pragma unroll
        for (int kk = 0; kk < 8; ++kk) {
          const int e0 = kk * 4 + 2 * hi;
          const int vc = ch0 + e0;
          v2f a, bm;
          a.x = Sh[arow * HD + e0];
          a.y = Sh[arow * HD + e0 + 1];
          bm.x = yln[vc * 64 + n] * wv[vc];
          bm.y = yln[(vc + 1) * 64 + n] * wv[vc + 1];
          acc = wmma_f32_16x16x4(a, bm, acc);
        }
#pragma unroll
        for (int v = 0; v < 8; ++v) {
          const int d = ti * 16 + v + 8 * hi;
          atomicAdd(&osum[h][d], acc[v]);   // ds_add_f32
        }
      }
    }
  }
  __syncthreads();

  // ---- c_att = sigmoid(mean over n)
  {
    const int c = t;
    const float o = osum[c >> 5][c & 31] * (1.0f / 64.0f);
    c_att[(size_t)b * CHN + c] = sigmoidf_(o);
  }
}

// ---------------------------------------------------------------------------
// Kernel 4: out = x * (h_att[h] * w_att[w] + c_att). Second streaming pass,
// x expected resident in the 192 MB L2.
// ---------------------------------------------------------------------------
__global__ __launch_bounds__(256)
void fuse_kernel(const float* __restrict__ x,
                 const float* __restrict__ h_att,
                 const float* __restrict__ w_att,
                 const float* __restrict__ c_att,
                 float* __restrict__ out) {
  __shared__ float ha[HH];
  __shared__ float wa[WW];
  const int plane = blockIdx.x;
  const int t = threadIdx.x;
  if (t < HH) {
    ha[t] = h_att[(size_t)plane * HH + t];
    wa[t] = w_att[(size_t)plane * WW + t];
  }
  __syncthreads();
  const float ca = c_att[plane];
  const float4* __restrict__ xp = (const float4*)(x + (size_t)plane * (HH * WW));
  float4* __restrict__ op = (float4*)(out + (size_t)plane * (HH * WW));
#pragma unroll
  for (int i = 0; i < 4; ++i) {
    const int idx4 = i * 256 + t;       // float4 index within plane
    const int hrow = idx4 >> 4;         // 16 float4 per row of 64
    const int w0 = (idx4 & 15) * 4;
    const float4 xv = xp[idx4];
    const float hv = ha[hrow];
    float4 r;
    r.x = xv.x * (hv * wa[w0 + 0] + ca);
    r.y = xv.y * (hv * wa[w0 + 1] + ca);
    r.z = xv.z * (hv * wa[w0 + 2] + ca);
    r.w = xv.w * (hv * wa[w0 + 3] + ca);
    op[idx4] = r;
  }
}

// ---------------------------------------------------------------------------
extern "C" void kernel_launch(void* const* d_in, const int* in_sizes, int n_in,
                              void* d_out, int out_size, void* d_ws, size_t ws_size,
                              hipStream_t stream) {
  const float* x      = (const float*)d_in[0];
  const float* dw1    = (const float*)d_in[1];
  const float* db1    = (const float*)d_in[2];
  const float* dw2    = (const float*)d_in[3];
  const float* db2    = (const float*)d_in[4];
  const float* dw3    = (const float*)d_in[5];
  const float* db3    = (const float*)d_in[6];
  const float* dw4    = (const float*)d_in[7];
  const float* db4    = (const float*)d_in[8];
  const float* gnh_w  = (const float*)d_in[9];
  const float* gnh_b  = (const float*)d_in[10];
  const float* gnw_w  = (const float*)d_in[11];
  const float* gnw_b  = (const float*)d_in[12];
  const float* norm_w = (const float*)d_in[13];
  const float* norm_b = (const float*)d_in[14];
  const float* wq     = (const float*)d_in[15];
  const float* wk     = (const float*)d_in[16];
  const float* wv     = (const float*)d_in[17];
  float* out = (float*)d_out;

  // workspace layout (floats)
  float* ws = (float*)d_ws;
  const size_t SEG = (size_t)PLANES * 64;   // 524288 floats per segment
  float* rowMean = ws;                      // (B,C,H)
  float* colMean = rowMean + SEG;           // (B,C,W)
  float* pooled  = colMean + SEG;           // (B,C,64)
  float* h_att   = pooled + SEG;            // (B,C,H)
  float* w_att   = h_att + SEG;             // (B,C,W)
  float* c_att   = w_att + SEG;             // (B,C)

  reduce_kernel<<<PLANES, 256, 0, stream>>>(x, rowMean, colMean, pooled);
  strip_kernel<<<dim3(BATCH, 2), 256, 0, stream>>>(
      rowMean, colMean, dw1, db1, dw2, db2, dw3, db3, dw4, db4,
      gnh_w, gnh_b, gnw_w, gnw_b, h_att, w_att);
  pool_att_kernel<<<BATCH, 256, 0, stream>>>(pooled, norm_w, norm_b,
                                             wq, wk, wv, c_att);
  fuse_kernel<<<PLANES, 256, 0, stream>>>(x, h_att, w_att, c_att, out);
}